// RRN_38843684225221
// MI455X (gfx1250) — compile-verified
//
#include <hip/hip_runtime.h>
#include <hip/hip_bf16.h>
#include <math.h>

typedef float v2f __attribute__((ext_vector_type(2)));
typedef float v8f __attribute__((ext_vector_type(8)));

// D = A(16x4 f32) * B(4x16 f32) + C(16x16 f32), full fp32 WMMA.
__device__ __forceinline__ v8f wmma4(v2f a, v2f b, v8f c) {
    return __builtin_amdgcn_wmma_f32_16x16x4_f32(
        /*neg_a=*/false, a, /*neg_b=*/false, b,
        /*c_mod=*/(short)0, c, /*reuse_a=*/false, /*reuse_b=*/false);
}

__device__ __forceinline__ v2f ld2(const float* p) { return *(const v2f*)p; }

__device__ __forceinline__ float sigmoidf_(float x) {
    return 1.0f / (1.0f + __expf(-x));
}

// ---- LDS strides padded so (stride % 64) == 4 dwords -> conflict-free b64 frags
#define LD128 132   // padded stride for K=128 rows
#define LD192 196   // padded stride for K=192 rows
#define LD64  68    // padded stride for K=64 rows

// ---------------------------------------------------------------------------
// Kernel 0: zero the segment-sum accumulator in workspace
// ---------------------------------------------------------------------------
__global__ void __launch_bounds__(256) rrn_zero_f4(float4* __restrict__ p, int n4) {
    int i = blockIdx.x * 256 + threadIdx.x;
    if (i < n4) p[i] = float4{0.f, 0.f, 0.f, 0.f};
}

// ---------------------------------------------------------------------------
// Kernel 1: edge MLP + fused atomic scatter-sum into sum_msgs [N,64]
// One wave processes 32 edges per tile (two 16-row M-tiles sharing every
// B fragment -> 2 wmma per LDS B-load). Layer1: 512 wmma, Layer2: 256 wmma
// per tile. Index loads pipelined one grid-stride tile ahead + row prefetch.
// ---------------------------------------------------------------------------
__global__ void __launch_bounds__(256) rrn_edge_msgs(
    const float* __restrict__ hidden,
    const int*   __restrict__ src, const int* __restrict__ dst,
    const float* __restrict__ fw1, const float* __restrict__ fb1,
    const float* __restrict__ fw2, const float* __restrict__ fb2,
    float* __restrict__ sum_msgs, int eTiles)   // eTiles = E/32
{
    extern __shared__ float smem[];
    float* sFw1T = smem;                       // [128 n][LD128 k]
    float* sFw2T = sFw1T + 128 * LD128;        // [64  n][LD128 k]
    float* sFb1  = sFw2T + 64 * LD128;         // [128]
    float* sFb2  = sFb1 + 128;                 // [64]
    float* sHmid = sFb2 + 64;                  // 8 waves * [32][LD128]

    const int tid = threadIdx.x;
    // stage fw1 (2H=128 x FH=128), transposed: T[n][k] = fw1[k][n]
    for (int i = tid; i < 128 * 128; i += 256) {
        int k = i >> 7, n = i & 127;
        sFw1T[n * LD128 + k] = fw1[i];
    }
    // stage fw2 (FH=128 x M=64), transposed
    for (int i = tid; i < 128 * 64; i += 256) {
        int k = i >> 6, n = i & 63;
        sFw2T[n * LD128 + k] = fw2[i];
    }
    if (tid < 128) sFb1[tid] = fb1[tid];
    if (tid < 64)  sFb2[tid] = fb2[tid];
    __syncthreads();

    const int w = tid >> 5, lane = tid & 31;
    const int col = lane & 15;      // N-column / A-row this lane owns
    const int hi  = lane >> 4;      // 0: K pair {k0,k0+1}, 1: {k0+2,k0+3}
    float* hmid = sHmid + w * (32 * LD128);

    const int gw = blockIdx.x * 8 + w;
    const int nw = gridDim.x * 8;

    // software-pipelined index loads for the first tile
    int s0n = 0, d0n = 0, s1n = 0, d1n = 0;
    if (gw < eTiles) {
        const int e0 = gw << 5;
        s0n = src[e0 + col];      d0n = dst[e0 + col];
        s1n = src[e0 + 16 + col]; d1n = dst[e0 + 16 + col];
    }

    for (int t = gw; t < eTiles; t += nw) {
        const int s0 = s0n, d0 = d0n, s1 = s1n, d1 = d1n;

        // preload next tile's indices and prefetch its gather rows
        const int tn = t + nw;
        if (tn < eTiles) {
            const int en = tn << 5;
            s0n = src[en + col];      d0n = dst[en + col];
            s1n = src[en + 16 + col]; d1n = dst[en + 16 + col];
            __builtin_prefetch(hidden + (size_t)s0n * 64, 0, 3);
            __builtin_prefetch(hidden + (size_t)s0n * 64 + 32, 0, 3);
            __builtin_prefetch(hidden + (size_t)d0n * 64, 0, 3);
            __builtin_prefetch(hidden + (size_t)d0n * 64 + 32, 0, 3);
            __builtin_prefetch(hidden + (size_t)s1n * 64, 0, 3);
            __builtin_prefetch(hidden + (size_t)s1n * 64 + 32, 0, 3);
            __builtin_prefetch(hidden + (size_t)d1n * 64, 0, 3);
            __builtin_prefetch(hidden + (size_t)d1n * 64 + 32, 0, 3);
        }

        // ---- layer 1: [32x128] (hidden[src]|hidden[dst]) @ fw1 -> [32x128]
        v8f acc0[8] = {};
        v8f acc1[8] = {};
        #pragma unroll 2
        for (int j = 0; j < 32; ++j) {
            const int klo = 4 * j + 2 * hi;
            const float* ap0 = (klo < 64)
                ? (hidden + (size_t)s0 * 64 + klo)
                : (hidden + (size_t)d0 * 64 + (klo - 64));
            const float* ap1 = (klo < 64)
                ? (hidden + (size_t)s1 * 64 + klo)
                : (hidden + (size_t)d1 * 64 + (klo - 64));
            v2f a0 = ld2(ap0);
            v2f a1 = ld2(ap1);
            #pragma unroll
            for (int nt = 0; nt < 8; ++nt) {
                v2f b = ld2(&sFw1T[(nt * 16 + col) * LD128 + klo]);
                acc0[nt] = wmma4(a0, b, acc0[nt]);
                acc1[nt] = wmma4(a1, b, acc1[nt]);
            }
        }

        // bias + ReLU, re-layout C->A via per-wave LDS staging (both M-tiles)
        #pragma unroll
        for (int nt = 0; nt < 8; ++nt) {
            const float bv = sFb1[nt * 16 + col];
            #pragma unroll
            for (int v = 0; v < 8; ++v) {
                const int m = v + 8 * hi;
                float v0 = acc0[nt][v] + bv;
                float v1 = acc1[nt][v] + bv;
                hmid[m * LD128 + nt * 16 + col]        = v0 > 0.0f ? v0 : 0.0f;
                hmid[(m + 16) * LD128 + nt * 16 + col] = v1 > 0.0f ? v1 : 0.0f;
            }
        }
        asm volatile("s_wait_dscnt 0" ::: "memory");  // LDS in-order per wave

        // ---- layer 2: [32x128] @ fw2 -> [32x64]
        v8f acc20[4] = {};
        v8f acc21[4] = {};
        #pragma unroll 2
        for (int j = 0; j < 32; ++j) {
            const int klo = 4 * j + 2 * hi;
            v2f a0 = ld2(&hmid[col * LD128 + klo]);
            v2f a1 = ld2(&hmid[(col + 16) * LD128 + klo]);
            #pragma unroll
            for (int nt = 0; nt < 4; ++nt) {
                v2f b = ld2(&sFw2T[(nt * 16 + col) * LD128 + klo]);
                acc20[nt] = wmma4(a0, b, acc20[nt]);
                acc21[nt] = wmma4(a1, b, acc21[nt]);
            }
        }

        // bias + scatter atomic-add by destination node (L2-resident target)
        #pragma unroll
        for (int nt = 0; nt < 4; ++nt) {
            const float bv = sFb2[nt * 16 + col];
            #pragma unroll
            for (int v = 0; v < 8; ++v) {
                const int m = v + 8 * hi;
                const int dr0 = __shfl(d0, m, 32);
                const int dr1 = __shfl(d1, m, 32);
                atomicAdd(&sum_msgs[(size_t)dr0 * 64 + nt * 16 + col],
                          acc20[nt][v] + bv);
                atomicAdd(&sum_msgs[(size_t)dr1 * 64 + nt * 16 + col],
                          acc21[nt][v] + bv);
            }
        }
    }
}

// ---------------------------------------------------------------------------
// Kernel 2: fused LSTM cell. gates[16x256] = [x|sum_msgs|h0](16x192) @ BT
// where BT[k][g] = k<128 ? Wih[g][k] : Whh[g][k-128]. 768 wmma / 16 nodes.
// ---------------------------------------------------------------------------
__global__ void __launch_bounds__(256) rrn_lstm(
    const float* __restrict__ x,  const float* __restrict__ sum_msgs,
    const float* __restrict__ h0, const float* __restrict__ c0,
    const float* __restrict__ Wih, const float* __restrict__ Whh,
    const float* __restrict__ bih, const float* __restrict__ bhh,
    float* __restrict__ h_new, float* __restrict__ c_new, int nTiles)
{
    extern __shared__ float smem[];
    float* sBT   = smem;                 // [256 g][LD192 k]
    float* sBias = sBT + 256 * LD192;    // [256] = bih+bhh

    const int tid = threadIdx.x;
    for (int i = tid; i < 256 * 192; i += 256) {
        const int g = i / 192, k = i % 192;
        sBT[g * LD192 + k] = (k < 128) ? Wih[g * 128 + k]
                                       : Whh[g * 64 + (k - 128)];
    }
    if (tid < 256) sBias[tid] = bih[tid] + bhh[tid];
    __syncthreads();

    const int w = tid >> 5, lane = tid & 31;
    const int col = lane & 15, hi = lane >> 4;
    const int gw = blockIdx.x * 8 + w, nw = gridDim.x * 8;

    for (int t = gw; t < nTiles; t += nw) {
        const int n0 = t << 4;
        const int nr = n0 + col;  // A-row node index for this lane

        v8f acc[16] = {};
        #pragma unroll 2
        for (int j = 0; j < 48; ++j) {
            const int klo = 4 * j + 2 * hi;
            const float* ap;
            if (klo < 64)       ap = x        + (size_t)nr * 64 + klo;
            else if (klo < 128) ap = sum_msgs + (size_t)nr * 64 + (klo - 64);
            else                ap = h0       + (size_t)nr * 64 + (klo - 128);
            v2f a = ld2(ap);
            #pragma unroll
            for (int g = 0; g < 16; ++g) {
                v2f b = ld2(&sBT[(g * 16 + col) * LD192 + klo]);
                acc[g] = wmma4(a, b, acc[g]);
            }
        }

        // Gate split (i|f|g|o over 64 cols each): tile ht of hidden col-range
        // uses acc[ht], acc[ht+4], acc[ht+8], acc[ht+12] at identical slots.
        #pragma unroll
        for (int ht = 0; ht < 4; ++ht) {
            const float bi = sBias[  0 + ht * 16 + col];
            const float bf = sBias[ 64 + ht * 16 + col];
            const float bg = sBias[128 + ht * 16 + col];
            const float bo = sBias[192 + ht * 16 + col];
            #pragma unroll
            for (int v = 0; v < 8; ++v) {
                const int m = v + 8 * hi;
                const size_t idx = (size_t)(n0 + m) * 64 + ht * 16 + col;
                const float ig = sigmoidf_(acc[ht     ][v] + bi);
                const float fg = sigmoidf_(acc[ht +  4][v] + bf);
                const float gg = tanhf    (acc[ht +  8][v] + bg);
                const float og = sigmoidf_(acc[ht + 12][v] + bo);
                const float cn = fg * c0[idx] + ig * gg;
                c_new[idx] = cn;
                h_new[idx] = og * tanhf(cn);
            }
        }
    }
}

// ---------------------------------------------------------------------------
// Kernel 3: output MLP: h_new[16x64] @ ow1 -> ReLU -> @ ow2 -> out[16x32]
// ---------------------------------------------------------------------------
__global__ void __launch_bounds__(256) rrn_out_mlp(
    const float* __restrict__ h_new,
    const float* __restrict__ ow1, const float* __restrict__ ob1,
    const float* __restrict__ ow2, const float* __restrict__ ob2,
    float* __restrict__ out, int nTiles)
{
    extern __shared__ float smem[];
    float* sW1T = smem;                   // [128 n][LD64 k]
    float* sW2T = sW1T + 128 * LD64;      // [32  n][LD128 k]
    float* sB1  = sW2T + 32 * LD128;      // [128]
    float* sB2  = sB1 + 128;              // [32]
    float* sHmid = sB2 + 32;              // 8 waves * [16][LD128]

    const int tid = threadIdx.x;
    for (int i = tid; i < 64 * 128; i += 256) {   // ow1 (64 x 128)
        int k = i >> 7, n = i & 127;
        sW1T[n * LD64 + k] = ow1[i];
    }
    for (int i = tid; i < 128 * 32; i += 256) {   // ow2 (128 x 32)
        int k = i >> 5, n = i & 31;
        sW2T[n * LD128 + k] = ow2[i];
    }
    if (tid < 128) sB1[tid] = ob1[tid];
    if (tid < 32)  sB2[tid] = ob2[tid];
    __syncthreads();

    const int w = tid >> 5, lane = tid & 31;
    const int col = lane & 15, hi = lane >> 4;
    float* hmid = sHmid + w * (16 * LD128);
    const int gw = blockIdx.x * 8 + w, nw = gridDim.x * 8;

    for (int t = gw; t < nTiles; t += nw) {
        const int n0 = t << 4;

        v8f acc[8] = {};
        #pragma unroll
        for (int j = 0; j < 16; ++j) {            // K = 64
            const int klo = 4 * j + 2 * hi;
            v2f a = ld2(&h_new[(size_t)(n0 + col) * 64 + klo]);
            #pragma unroll
            for (int nt = 0; nt < 8; ++nt) {
                v2f b = ld2(&sW1T[(nt * 16 + col) * LD64 + klo]);
                acc[nt] = wmma4(a, b, acc[nt]);
            }
        }
        #pragma unroll
        for (int nt = 0; nt < 8; ++nt) {
            const float bv = sB1[nt * 16 + col];
            #pragma unroll
            for (int v = 0; v < 8; ++v) {
                const int m = v + 8 * hi;
                float val = acc[nt][v] + bv;
                hmid[m * LD128 + nt * 16 + col] = val > 0.0f ? val : 0.0f;
            }
        }
        asm volatile("s_wait_dscnt 0" ::: "memory");

        v8f acc2[2] = {};
        #pragma unroll 4
        for (int j = 0; j < 32; ++j) {            // K = 128
            const int klo = 4 * j + 2 * hi;
            v2f a = ld2(&hmid[col * LD128 + klo]);
            #pragma unroll
            for (int nt = 0; nt < 2; ++nt) {
                v2f b = ld2(&sW2T[(nt * 16 + col) * LD128 + klo]);
                acc2[nt] = wmma4(a, b, acc2[nt]);
            }
        }
        #pragma unroll
        for (int nt = 0; nt < 2; ++nt) {
            const float bv = sB2[nt * 16 + col];
            #pragma unroll
            for (int v = 0; v < 8; ++v) {
                const int m = v + 8 * hi;
                out[(size_t)(n0 + m) * 32 + nt * 16 + col] = acc2[nt][v] + bv;
            }
        }
    }
}

// ---------------------------------------------------------------------------
extern "C" void kernel_launch(void* const* d_in, const int* in_sizes, int n_in,
                              void* d_out, int out_size, void* d_ws, size_t ws_size,
                              hipStream_t stream) {
    const float* x      = (const float*)d_in[0];
    const float* hidden = (const float*)d_in[1];
    const float* h0     = (const float*)d_in[2];
    const float* c0     = (const float*)d_in[3];
    const int*   src    = (const int*)d_in[4];
    const int*   dst    = (const int*)d_in[5];
    const float* fw1    = (const float*)d_in[6];
    const float* fb1    = (const float*)d_in[7];
    const float* fw2    = (const float*)d_in[8];
    const float* fb2    = (const float*)d_in[9];
    const float* Wih    = (const float*)d_in[10];
    const float* Whh    = (const float*)d_in[11];
    const float* bih    = (const float*)d_in[12];
    const float* bhh    = (const float*)d_in[13];
    const float* ow1    = (const float*)d_in[14];
    const float* ob1    = (const float*)d_in[15];
    const float* ow2    = (const float*)d_in[16];
    const float* ob2    = (const float*)d_in[17];

    const int N = in_sizes[0] / 64;   // 50000
    const int E = in_sizes[4];        // 1600000

    float* sum_msgs = (float*)d_ws;                       // [N,64]
    float* out   = (float*)d_out;                         // [N,32]
    float* h_new = out + (size_t)N * 32;                  // [N,64]
    float* c_new = h_new + (size_t)N * 64;                // [N,64]

    // 0) zero the scatter accumulator
    const int n4 = (N * 64) / 4;
    rrn_zero_f4<<<(n4 + 255) / 256, 256, 0, stream>>>((float4*)sum_msgs, n4);

    // 1) edge MLP + scatter (78.6 GFLOP, dominant) -- 32 edges per wave-tile
    const int eTiles = E / 32;
    const size_t shEdge =
        (size_t)(128 * LD128 + 64 * LD128 + 128 + 64 + 8 * 32 * LD128) * sizeof(float);
    rrn_edge_msgs<<<1024, 256, shEdge, stream>>>(
        hidden, src, dst, fw1, fb1, fw2, fb2, sum_msgs, eTiles);

    // 2) LSTM cell
    const int nTiles = N / 16;
    const size_t shLstm = (size_t)(256 * LD192 + 256) * sizeof(float);
    rrn_lstm<<<256, 256, shLstm, stream>>>(
        x, sum_msgs, h0, c0, Wih, Whh, bih, bhh, h_new, c_new, nTiles);

    // 3) output MLP
    const size_t shOut =
        (size_t)(128 * LD64 + 32 * LD128 + 128 + 32 + 8 * 16 * LD128) * sizeof(float);
    rrn_out_mlp<<<256, 256, shOut, stream>>>(
        h_new, ow1, ob1, ow2, ob2, out, nTiles);
}